// AggregationAttention_17910013624836
// MI455X (gfx1250) — compile-verified
//
#include <hip/hip_runtime.h>
#include <hip/hip_bf16.h>
#include <math.h>

typedef __attribute__((ext_vector_type(2))) float v2f;
typedef __attribute__((ext_vector_type(8))) float v8f;

// Problem constants
#define BB 8
#define SS 4096
#define EE 1024
#define HH 16
#define ROWS (BB * SS)          // 32768
#define HID (4 * EE)            // 4096
#define OUTD 1024

// Workspace layout (float offsets)
#define WS_Q      0                      // 1024
#define WS_QKT    1024                   // 16384 : qkT[e][h] (pre-scaled by 1/8)
#define WS_QB     (WS_QKT + 16384)       // 16
#define WS_LOGITS (WS_QB + 16)           // 524288 : [B][H][S]
#define WS_PROBS  (WS_LOGITS + 524288)   // 524288 : A * softmax
#define WS_AMASK  (WS_PROBS + 524288)    // 524288
#define WS_SUMP   (WS_AMASK + 524288)    // 128
#define WS_CTX    (WS_SUMP + 128)        // 131072 : [B][H][E]
#define WS_HN     (WS_CTX + 131072)      // 8192   : normalized hidden
#define WS_H1     (WS_HN + 8192)         // 32768  : relu(hidden@W1.T+b1)

// ---------------------------------------------------------------------------
// Kernel A: q = query@Wq.T+bq ; qk[h] folded through Wk ; qb[h] ; scale 1/8
// ---------------------------------------------------------------------------
__global__ void k_setup(const float* __restrict__ query, const float* __restrict__ Wq,
                        const float* __restrict__ bq, const float* __restrict__ Wk,
                        const float* __restrict__ bk, float* __restrict__ ws) {
  __shared__ float qsh[EE];
  const int t = threadIdx.x;
  for (int j = t; j < EE; j += 256) {
    float acc = 0.f;
    const float* w = Wq + (size_t)j * EE;
    for (int e = 0; e < EE; ++e) acc += query[e] * w[e];
    acc += bq[j];
    qsh[j] = acc;
    ws[WS_Q + j] = acc;
  }
  __syncthreads();
  // qkT[e*16+h] = 0.125 * sum_d q[h*64+d] * Wk[(h*64+d)*E + e]
  for (int idx = t; idx < EE * HH; idx += 256) {
    int e = idx >> 4, h = idx & 15;
    float acc = 0.f;
    for (int d = 0; d < 64; ++d) acc += qsh[h * 64 + d] * Wk[(size_t)(h * 64 + d) * EE + e];
    ws[WS_QKT + idx] = 0.125f * acc;
  }
  if (t < HH) {
    float acc = 0.f;
    for (int d = 0; d < 64; ++d) acc += qsh[t * 64 + d] * bk[t * 64 + d];
    ws[WS_QB + t] = 0.125f * acc;
  }
}

// ---------------------------------------------------------------------------
// Kernel B: logits[b,h,s] = x @ qkT  via V_WMMA_F32_16X16X4_F32
// 256 threads = 8 waves; each wave owns a 16-row tile; x staged in LDS.
// Grid = ROWS/128 = 256 blocks.
// ---------------------------------------------------------------------------
__global__ void k_logits(const float* __restrict__ x, const float* __restrict__ qkT,
                         const float* __restrict__ qb_arr, float* __restrict__ logits) {
  __shared__ float xs[128][65];   // +1 pad: 260B row stride breaks 256B bank period
  const int t = threadIdx.x;
  const int wave = t >> 5, lane = t & 31;
  const int rowBlock = blockIdx.x * 128;
  const int m = lane & 15;
  const int koff = (lane < 16) ? 0 : 2;
  v8f acc = {};
  for (int e0 = 0; e0 < EE; e0 += 64) {
    for (int l = t; l < 128 * 64; l += 256) {
      int r = l >> 6, c = l & 63;
      xs[r][c] = x[(size_t)(rowBlock + r) * EE + e0 + c];
    }
    __syncthreads();
    const int tr = wave * 16;
    for (int k = 0; k < 64; k += 4) {
      v2f a, b;
      a.x = xs[tr + m][k + koff];
      a.y = xs[tr + m][k + koff + 1];
      const int eg = e0 + k + koff;
      b.x = qkT[(size_t)eg * HH + m];
      b.y = qkT[(size_t)(eg + 1) * HH + m];
      acc = __builtin_amdgcn_wmma_f32_16x16x4_f32(false, a, false, b, (short)0, acc,
                                                  false, false);
    }
    __syncthreads();
  }
  const float qb = qb_arr[m];          // N = head = lane%16
  for (int i = 0; i < 8; ++i) {
    int row = rowBlock + wave * 16 + i + ((lane < 16) ? 0 : 8);   // M
    int bidx = row >> 12, s = row & 4095;
    logits[((size_t)(bidx * HH + m)) * SS + s] = acc[i] + qb;
  }
}

// ---------------------------------------------------------------------------
// Kernel C: per (b,h): softmax over S, gumbel hard mask, masked probs, sumP
// Grid = B*H = 128 blocks, 256 threads.
// ---------------------------------------------------------------------------
__global__ void k_softmax_gumbel(const float* __restrict__ logits, const float* __restrict__ gum,
                                 float* __restrict__ probs, float* __restrict__ Amask,
                                 float* __restrict__ sumP) {
  __shared__ float red[256];
  const int bh = blockIdx.x, t = threadIdx.x;
  const float* L = logits + (size_t)bh * SS;
  float mx = -INFINITY;
  for (int s = t; s < SS; s += 256) mx = fmaxf(mx, L[s]);
  red[t] = mx; __syncthreads();
  for (int o = 128; o > 0; o >>= 1) { if (t < o) red[t] = fmaxf(red[t], red[t + o]); __syncthreads(); }
  mx = red[0]; __syncthreads();
  float sm = 0.f;
  for (int s = t; s < SS; s += 256) sm += __expf(L[s] - mx);
  red[t] = sm; __syncthreads();
  for (int o = 128; o > 0; o >>= 1) { if (t < o) red[t] += red[t + o]; __syncthreads(); }
  const float inv = 1.0f / red[0]; __syncthreads();
  float sp = 0.f;
  for (int s = t; s < SS; s += 256) {
    const float l = L[s];
    const float u0 = gum[((size_t)bh * SS + s) * 2 + 0];
    const float u1 = gum[((size_t)bh * SS + s) * 2 + 1];
    const float g0 = -__logf(-__logf(u0 + 1e-20f) + 1e-20f);
    const float g1 = -__logf(-__logf(u1 + 1e-20f) + 1e-20f);
    const float A = (l + g1 > g0) ? 1.0f : 0.0f;   // hard straight-through forward
    const float p = A * __expf(l - mx) * inv;
    probs[(size_t)bh * SS + s] = p;
    Amask[(size_t)bh * SS + s] = A;
    sp += p;
  }
  red[t] = sp; __syncthreads();
  for (int o = 128; o > 0; o >>= 1) { if (t < o) red[t] += red[t + o]; __syncthreads(); }
  if (t == 0) sumP[bh] = red[0];
}

// ---------------------------------------------------------------------------
// Kernel D: ctx[b,h,e] = sum_s probs[b,h,s] * x[b,s,e]   (WMMA, A=probs B=x)
// Grid = B*64 blocks (b, 16-wide e-tile); 8 waves split S; LDS cross-wave sum.
// ---------------------------------------------------------------------------
__global__ void k_ctx(const float* __restrict__ x, const float* __restrict__ probs,
                      float* __restrict__ ctx) {
  __shared__ float red[8][256];
  const int b = blockIdx.x >> 6;
  const int e0 = (blockIdx.x & 63) * 16;
  const int t = threadIdx.x, wave = t >> 5, lane = t & 31;
  const int m = lane & 15;
  const int koff = (lane < 16) ? 0 : 2;
  const float* Pb = probs + (size_t)b * HH * SS;
  const float* Xb = x + (size_t)b * SS * EE;
  v8f acc = {};
  const int sEnd = (wave + 1) * 512;
  for (int s = wave * 512; s < sEnd; s += 4) {
    v2f a, bm;
    a.x = Pb[(size_t)m * SS + s + koff];          // A[M=head][K=s]
    a.y = Pb[(size_t)m * SS + s + koff + 1];
    bm.x = Xb[(size_t)(s + koff) * EE + e0 + m];  // B[K=s][N=e]
    bm.y = Xb[(size_t)(s + koff + 1) * EE + e0 + m];
    acc = __builtin_amdgcn_wmma_f32_16x16x4_f32(false, a, false, bm, (short)0, acc,
                                                false, false);
  }
  for (int i = 0; i < 8; ++i) red[wave][i * 32 + lane] = acc[i];
  __syncthreads();
  float ssum = 0.f;
  for (int w = 0; w < 8; ++w) ssum += red[w][t];
  const int i = t >> 5, l = t & 31;
  const int h = i + ((l & 16) ? 8 : 0);   // M
  const int n = l & 15;                   // N
  ctx[((size_t)(b * HH + h)) * EE + e0 + n] = ssum;
}

// ---------------------------------------------------------------------------
// Kernel E: hidden = Wv_h @ ctx + sumP*bv ; layernorm.  Grid = B blocks.
// ---------------------------------------------------------------------------
__global__ void k_hidden_ln(const float* __restrict__ Wv, const float* __restrict__ bv,
                            const float* __restrict__ ctx, const float* __restrict__ sumP,
                            const float* __restrict__ ln_g, const float* __restrict__ ln_b,
                            float* __restrict__ hn) {
  __shared__ float hsh[EE];
  __shared__ float red[256];
  const int b = blockIdx.x, t = threadIdx.x;
  for (int j = t; j < EE; j += 256) {
    const int h = j >> 6;
    const float* c = ctx + (size_t)(b * HH + h) * EE;
    const float* w = Wv + (size_t)j * EE;
    float acc = 0.f;
    for (int e = 0; e < EE; ++e) acc += w[e] * c[e];
    hsh[j] = acc + sumP[b * HH + h] * bv[j];
  }
  __syncthreads();
  float lm = 0.f;
  for (int j = t; j < EE; j += 256) lm += hsh[j];
  red[t] = lm; __syncthreads();
  for (int o = 128; o > 0; o >>= 1) { if (t < o) red[t] += red[t + o]; __syncthreads(); }
  const float mu = red[0] * (1.0f / EE); __syncthreads();
  float lv = 0.f;
  for (int j = t; j < EE; j += 256) { float d = hsh[j] - mu; lv += d * d; }
  red[t] = lv; __syncthreads();
  for (int o = 128; o > 0; o >>= 1) { if (t < o) red[t] += red[t + o]; __syncthreads(); }
  const float rstd = rsqrtf(red[0] * (1.0f / EE) + 1e-5f);
  for (int j = t; j < EE; j += 256)
    hn[(size_t)b * EE + j] = (hsh[j] - mu) * rstd * ln_g[j] + ln_b[j];
}

// ---------------------------------------------------------------------------
// Kernel G: h1 = relu(hn @ W1.T + b1). Grid = HID/256; W1 streamed once.
// ---------------------------------------------------------------------------
__global__ void k_mlp1(const float* __restrict__ hn, const float* __restrict__ W1,
                       const float* __restrict__ b1, float* __restrict__ h1) {
  __shared__ float hs[BB][EE];
  const int t = threadIdx.x;
  for (int l = t; l < BB * EE; l += 256) hs[l >> 10][l & 1023] = hn[l];
  __syncthreads();
  const int mrow = blockIdx.x * 256 + t;
  const float* w = W1 + (size_t)mrow * EE;
  float acc[BB] = {};
  for (int e = 0; e < EE; ++e) {
    const float wv = w[e];
    for (int b = 0; b < BB; ++b) acc[b] += wv * hs[b][e];
  }
  const float bias = b1[mrow];
  for (int b = 0; b < BB; ++b)
    h1[(size_t)b * HID + mrow] = fmaxf(acc[b] + bias, 0.f);
}

// ---------------------------------------------------------------------------
// Kernel H: out = h1 @ W2.T + b2. Grid = OUTD/256; W2 streamed once, h1 in L2.
// ---------------------------------------------------------------------------
__global__ void k_mlp2(const float* __restrict__ h1, const float* __restrict__ W2,
                       const float* __restrict__ b2, float* __restrict__ out) {
  const int o = blockIdx.x * 256 + threadIdx.x;
  const float* w = W2 + (size_t)o * HID;
  float acc[BB] = {};
  for (int m = 0; m < HID; ++m) {
    const float wv = w[m];
    for (int b = 0; b < BB; ++b) acc[b] += wv * h1[(size_t)b * HID + m];
  }
  for (int b = 0; b < BB; ++b) out[(size_t)b * OUTD + o] = acc[b] + b2[o];
}

// ---------------------------------------------------------------------------
// Kernel M: masks/attn = head-sums. Grid = B*S/256.
// ---------------------------------------------------------------------------
__global__ void k_masks(const float* __restrict__ probs, const float* __restrict__ Amask,
                        float* __restrict__ masks, float* __restrict__ attn) {
  const int idx = blockIdx.x * 256 + threadIdx.x;  // b*S + s
  const int b = idx >> 12, s = idx & 4095;
  float ms = 0.f, at = 0.f;
  for (int h = 0; h < HH; ++h) {
    ms += Amask[((size_t)(b * HH + h)) * SS + s];
    at += probs[((size_t)(b * HH + h)) * SS + s];
  }
  masks[idx] = ms;
  attn[idx] = at;
}

// ---------------------------------------------------------------------------
extern "C" void kernel_launch(void* const* d_in, const int* in_sizes, int n_in,
                              void* d_out, int out_size, void* d_ws, size_t ws_size,
                              hipStream_t stream) {
  const float* x     = (const float*)d_in[0];
  const float* gum   = (const float*)d_in[1];
  const float* query = (const float*)d_in[2];
  const float* Wq    = (const float*)d_in[3];
  const float* bq    = (const float*)d_in[4];
  const float* Wk    = (const float*)d_in[5];
  const float* bk    = (const float*)d_in[6];
  const float* Wv    = (const float*)d_in[7];
  const float* bv    = (const float*)d_in[8];
  const float* ln_g  = (const float*)d_in[9];
  const float* ln_b  = (const float*)d_in[10];
  const float* W1    = (const float*)d_in[11];
  const float* b1    = (const float*)d_in[12];
  const float* W2    = (const float*)d_in[13];
  const float* b2    = (const float*)d_in[14];

  float* ws = (float*)d_ws;
  float* out = (float*)d_out;                 // [0, 8192): out
  float* masks = out + BB * OUTD;             // [8192, 40960): masks
  float* attn  = masks + BB * SS;             // [40960, 73728): attn

  k_setup<<<1, 256, 0, stream>>>(query, Wq, bq, Wk, bk, ws);
  k_logits<<<ROWS / 128, 256, 0, stream>>>(x, ws + WS_QKT, ws + WS_QB, ws + WS_LOGITS);
  k_softmax_gumbel<<<BB * HH, 256, 0, stream>>>(ws + WS_LOGITS, gum, ws + WS_PROBS,
                                                ws + WS_AMASK, ws + WS_SUMP);
  k_ctx<<<BB * 64, 256, 0, stream>>>(x, ws + WS_PROBS, ws + WS_CTX);
  k_hidden_ln<<<BB, 256, 0, stream>>>(Wv, bv, ws + WS_CTX, ws + WS_SUMP, ln_g, ln_b,
                                      ws + WS_HN);
  k_mlp1<<<HID / 256, 256, 0, stream>>>(ws + WS_HN, W1, b1, ws + WS_H1);
  k_mlp2<<<OUTD / 256, 256, 0, stream>>>(ws + WS_H1, W2, b2, out);
  k_masks<<<(BB * SS) / 256, 256, 0, stream>>>(ws + WS_PROBS, ws + WS_AMASK, masks, attn);
}